// NNDPF40DenseNN_50225347559741
// MI455X (gfx1250) — compile-verified
//
#include <hip/hip_runtime.h>
#include <math.h>

typedef float v2f __attribute__((ext_vector_type(2)));
typedef float v4f __attribute__((ext_vector_type(4)));
typedef float v8f __attribute__((ext_vector_type(8)));

#define IN_DIM 2
#define H1N    25
#define H2N    20
#define OUTN   700
#define NT     44   // 44 tiles of 16 cols cover 704 >= 700
#define KSN    5    // 5 k-steps of 4 cover K=20
#define WAVES  8
#define BLOCK  256

// ---------------------------------------------------------------------------
// One block = 256 samples (8 waves x 32 lanes; one sample per lane for the
// small layers). Layer 3 (20 -> 700) runs on V_WMMA_F32_16X16X4_F32 with
// double-buffered B fragments and two interleaved accumulator chains.
// The 4 trapz norms are linear in o, so they fold into 20-vectors
// g = W3[:,slice] @ trapz_w: each lane gets its scale factors from 80 FMAs.
// Per-column {b3, scale-slot} live in one 8-byte LDS record (single
// ds_load_b64 per tile); scales are slot-major with a 5th identity slot so
// the (d + b3) * sc store stream is completely branch-free. Only the final
// tile (cols 688..703) carries a store guard.
// ---------------------------------------------------------------------------
__global__ __launch_bounds__(BLOCK) void dense_nn_sumrules_kernel(
    const float* __restrict__ x,
    const float* __restrict__ W1, const float* __restrict__ b1,
    const float* __restrict__ W2, const float* __restrict__ b2,
    const float* __restrict__ W3, const float* __restrict__ b3,
    float* __restrict__ out)
{
    __shared__ float lds_B[NT * KSN * 2 * 32];            // W3 as B fragments
    __shared__ __align__(8) float lds_ci[NT * 16 * 2];    // {b3[c], slot*128}
    __shared__ float lds_w[50];                           // trapz weights
    __shared__ float lds_g[4 * H2N];                      // reduced norm vecs
    __shared__ float lds_gb[4];                           // b3 part of norms
    __shared__ float lds_h2[WAVES][32 * 21];              // stride 21
    __shared__ __align__(16) float lds_sc[WAVES][5][32];  // slot-major scales

    const int tid  = threadIdx.x;
    const int wave = tid >> 5;
    const int lane = tid & 31;
    const int lm   = lane & 15;
    const int lhK  = (lane >= 16) ? 2 : 0;   // K offset for hi half-wave (A)
    const int lhM  = (lane >= 16) ? 8 : 0;   // M offset for hi half-wave (C/D)

    // ---- stage trapezoid weights: XGRID = logspace(-7, 0, 50) --------------
    if (tid < 50) {
        const int i = tid;
        const float xi = exp10f(-7.0f + 7.0f * (float)i / 49.0f);
        float wv = 0.0f;
        if (i < 49) wv += 0.5f * (exp10f(-7.0f + 7.0f * (float)(i + 1) / 49.0f) - xi);
        if (i > 0)  wv += 0.5f * (xi - exp10f(-7.0f + 7.0f * (float)(i - 1) / 49.0f));
        lds_w[i] = wv;
    }
    // ---- stage per-column info: b3 value + scale-slot byte offset ----------
    for (int c = tid; c < NT * 16; c += BLOCK) {
        int slot = 4;                                    // 4 = identity scale
        if (c >= 50 && c < 300) slot = (c < 150) ? 0 : (c - 100) / 50;
        lds_ci[2 * c] = (c < OUTN) ? b3[c] : 0.0f;
        ((int*)lds_ci)[2 * c + 1] = slot * (32 * 4);     // byte offset of row
    }
    // ---- stage W3 in B-fragment layout for V_WMMA_F32_16X16X4_F32 ----------
    // VGPR0: rows K=4ks+0 (lanes 0-15), K=4ks+2 (lanes 16-31); VGPR1: +1/+3.
    for (int idx = tid; idx < NT * KSN * 2 * 32; idx += BLOCK) {
        const int l  = idx & 31;
        int rest     = idx >> 5;
        const int vg = rest & 1;  rest >>= 1;
        const int ks = rest % KSN;
        const int t  = rest / KSN;
        const int row = 4 * ks + vg + ((l >= 16) ? 2 : 0);
        const int col = t * 16 + (l & 15);
        lds_B[idx] = (col < OUTN) ? W3[row * OUTN + col] : 0.0f;
    }
    __syncthreads();

    // ---- reduced norm vectors: g[n][k] = sum_c w[c] * W3[k][slice_n(c)] ----
    if (tid < 4 * H2N) {
        const int norm = tid / H2N, k = tid % H2N;
        const float* wr = W3 + k * OUTN;
        float acc = 0.0f;
        if (norm == 0) {                 // g_sigma: trapz(o[50:100]+o[100:150])
            for (int i = 0; i < 50; ++i) acc += lds_w[i] * (wr[50 + i] + wr[100 + i]);
        } else {                         // v/v3/v8 at 150/200/250
            const int base = 100 + 50 * norm;
            for (int i = 0; i < 50; ++i) acc += lds_w[i] * wr[base + i];
        }
        lds_g[tid] = acc;
    } else if (tid < 4 * H2N + 4) {
        const int norm = tid - 4 * H2N;
        float acc = 0.0f;
        if (norm == 0) {
            for (int i = 0; i < 50; ++i) acc += lds_w[i] * (b3[50 + i] + b3[100 + i]);
        } else {
            const int base = 100 + 50 * norm;
            for (int i = 0; i < 50; ++i) acc += lds_w[i] * b3[base + i];
        }
        lds_gb[norm] = acc;
    }
    __syncthreads();

    // ---- layers 1 & 2, one sample per lane (VALU + v_tanh) -----------------
    const long long s = (long long)blockIdx.x * BLOCK + tid;
    const float x0 = x[2 * s], x1 = x[2 * s + 1];
    float h1v[H1N];
    #pragma unroll
    for (int j = 0; j < H1N; ++j)
        h1v[j] = tanhf(x0 * W1[j] + x1 * W1[H1N + j] + b1[j]);
    float h2v[H2N];
    #pragma unroll
    for (int k = 0; k < H2N; ++k) {
        float acc = b2[k];
        #pragma unroll
        for (int j = 0; j < H1N; ++j) acc += h1v[j] * W2[j * H2N + k];
        h2v[k] = tanhf(acc);
    }
    #pragma unroll
    for (int k = 0; k < H2N; ++k) lds_h2[wave][lane * 21 + k] = h2v[k];

    // ---- per-sample norms & scale factors (slot-major, slot 4 = identity) --
    float ns[4];
    #pragma unroll
    for (int n = 0; n < 4; ++n) {
        float acc = lds_gb[n];
        #pragma unroll
        for (int k = 0; k < H2N; ++k) acc += h2v[k] * lds_g[n * H2N + k];
        ns[n] = acc;
    }
    lds_sc[wave][0][lane] = 1.0f / ns[0];
    lds_sc[wave][1][lane] = 3.0f / ns[1];
    lds_sc[wave][2][lane] = 1.0f / ns[2];
    lds_sc[wave][3][lane] = 3.0f / ns[3];
    lds_sc[wave][4][lane] = 1.0f;

    // ---- A fragments: 16x4 fp32, lanes 0-15 K=4ks+{0,1}, 16-31 K=4ks+{2,3} -
    v2f afr[2][KSN];
    #pragma unroll
    for (int mt = 0; mt < 2; ++mt)
        #pragma unroll
        for (int ks = 0; ks < KSN; ++ks) {
            const int rb = (mt * 16 + lm) * 21 + 4 * ks + lhK;
            afr[mt][ks].x = lds_h2[wave][rb + 0];
            afr[mt][ks].y = lds_h2[wave][rb + 1];
        }

    const long long rowbase = (long long)blockIdx.x * BLOCK + wave * 32;
    float* const outp = out + (rowbase + lhM) * (long long)OUTN;
    const char* const scbase = (const char*)&lds_sc[wave][0][0] + lhM * 4;

    auto loadB = [&](int t, v2f* bfr) {
        #pragma unroll
        for (int ks = 0; ks < KSN; ++ks) {
            const int base = (t * KSN + ks) * 2 * 32 + lane;
            bfr[ks].x = lds_B[base];
            bfr[ks].y = lds_B[base + 32];
        }
    };

    auto processTile = [&](int t, const v2f* bfr, bool guard) {
        v8f d0 = {}, d1 = {};
        #pragma unroll
        for (int ks = 0; ks < KSN; ++ks) {   // two independent WMMA chains
            d0 = __builtin_amdgcn_wmma_f32_16x16x4_f32(
                    false, afr[0][ks], false, bfr[ks], (short)0, d0, false, false);
            d1 = __builtin_amdgcn_wmma_f32_16x16x4_f32(
                    false, afr[1][ks], false, bfr[ks], (short)0, d1, false, false);
        }
        const int c = t * 16 + lm;                       // my output column
        // one ds_load_b64: {b3[c], slot byte offset}
        struct CI { float b3; int off; };
        const CI ci = *(const CI*)__builtin_assume_aligned(&lds_ci[2 * c], 8);
        const float* scp = (const float*)(scbase + ci.off);
        const v4f sc0 = *(const v4f*)__builtin_assume_aligned(scp, 16);      // mt0 j=0..3
        const v4f sc1 = *(const v4f*)__builtin_assume_aligned(scp + 4, 16);  // mt0 j=4..7
        const v4f sc2 = *(const v4f*)__builtin_assume_aligned(scp + 16, 16); // mt1 j=0..3
        const v4f sc3 = *(const v4f*)__builtin_assume_aligned(scp + 20, 16); // mt1 j=4..7
        float* op = outp + c;
        if (!guard || c < OUTN) {
            #pragma unroll
            for (int j = 0; j < 4; ++j) op[j * OUTN]        = (d0[j]     + ci.b3) * sc0[j];
            #pragma unroll
            for (int j = 0; j < 4; ++j) op[(4 + j) * OUTN]  = (d0[4 + j] + ci.b3) * sc1[j];
            #pragma unroll
            for (int j = 0; j < 4; ++j) op[(16 + j) * OUTN] = (d1[j]     + ci.b3) * sc2[j];
            #pragma unroll
            for (int j = 0; j < 4; ++j) op[(20 + j) * OUTN] = (d1[4 + j] + ci.b3) * sc3[j];
        }
    };

    // ---- layer 3: software-pipelined over 44 N-tiles -----------------------
    v2f bcur[KSN], bnxt[KSN];
    loadB(0, bcur);
    for (int t = 0; t < NT - 1; ++t) {
        loadB(t + 1, bnxt);                  // prefetch next tile's B frags
        processTile(t, bcur, false);         // all columns valid: no guards
        #pragma unroll
        for (int ks = 0; ks < KSN; ++ks) bcur[ks] = bnxt[ks];
    }
    processTile(NT - 1, bcur, true);         // cols 688..703: single guard
}

extern "C" void kernel_launch(void* const* d_in, const int* in_sizes, int n_in,
                              void* d_out, int out_size, void* d_ws, size_t ws_size,
                              hipStream_t stream) {
    (void)n_in; (void)out_size; (void)d_ws; (void)ws_size;
    const float* x  = (const float*)d_in[0];
    const float* W1 = (const float*)d_in[1];
    const float* b1 = (const float*)d_in[2];
    const float* W2 = (const float*)d_in[3];
    const float* b2 = (const float*)d_in[4];
    const float* W3 = (const float*)d_in[5];
    const float* b3 = (const float*)d_in[6];
    float* out = (float*)d_out;

    const int batch  = in_sizes[0] / IN_DIM;   // 131072
    const int blocks = batch / BLOCK;          // 512 (batch divisible by 256)
    dense_nn_sumrules_kernel<<<blocks, BLOCK, 0, stream>>>(
        x, W1, b1, W2, b2, W3, b3, out);
}